// Model_36215164240759
// MI455X (gfx1250) — compile-verified
//
#include <hip/hip_runtime.h>
#include <math.h>

typedef __attribute__((ext_vector_type(2))) float v2f;
typedef __attribute__((ext_vector_type(8))) float v8f;
typedef __attribute__((ext_vector_type(4))) unsigned int v4u;
typedef __attribute__((ext_vector_type(4))) int v4i;
typedef __attribute__((ext_vector_type(8))) int v8i;

#define NNODES 100000
#define NEDGES 1600000
#define CH 40
#define KITER 10
#define LAM 0.5f

#define ROWS 64        // rows per MLP block
#define KSLAB 20       // K staged per slab
#define NSLAB 25       // 500 / 20
#define H1S 260        // LDS stride for hidden tile (256 + 4: conflict-free)
#define W1ROW 544      // LDS dword stride of one K-pair row (512 + 2x16 pad): 544%64==32
#define W2ROW 96       // LDS/global dword stride of one W2 K-pair row: 96%64==32

#if defined(__has_builtin)
#if __has_builtin(__builtin_amdgcn_tensor_load_to_lds) && __has_builtin(__builtin_amdgcn_s_wait_tensorcnt)
#define HAVE_TDM 1
#if __has_include(<hip/amd_detail/amd_gfx1250_TDM.h>)
#define TDM_6ARG 1
#endif
#endif
#endif

#ifdef HAVE_TDM
// 2-D tensor tile -> LDS via the Tensor Data Mover (D# per ISA sec 8.3/8.4).
// g1w0x: extra bits for group1 word0 (pad_enable / pad_interval / pad_amount).
__device__ __forceinline__ void tdm_load_2d(unsigned int lds_off, const void* gptr,
                                            unsigned int td0, unsigned int td1,
                                            unsigned long long stride0,
                                            unsigned int tile0, unsigned int tile1,
                                            unsigned int g1w0x) {
  unsigned long long ga = (unsigned long long)gptr;
  v4u g0;
  g0.x = 1u;                                                   // count=1, user mode
  g0.y = lds_off;                                              // lds_addr (bytes)
  g0.z = (unsigned int)(ga & 0xFFFFFFFFu);                     // global_addr[31:0]
  g0.w = (unsigned int)((ga >> 32) & 0x1FFFFFFu) | (2u << 30); // global_addr[56:32] | type=2
  v8i g1;
  g1[0] = (int)((2u << 16) | g1w0x);                           // data_size=4B | pad cfg
  g1[1] = (int)((td0 & 0xFFFFu) << 16);                        // tensor_dim0[15:0]
  g1[2] = (int)(((td0 >> 16) & 0xFFFFu) | ((td1 & 0xFFFFu) << 16));
  g1[3] = (int)(((td1 >> 16) & 0xFFFFu) | ((tile0 & 0xFFFFu) << 16));
  g1[4] = (int)(tile1 & 0xFFFFu);                              // tile_dim1; tile_dim2=0
  g1[5] = (int)(unsigned int)(stride0 & 0xFFFFFFFFull);
  g1[6] = (int)(unsigned int)((stride0 >> 32) & 0xFFFFull);    // dim1_stride = 0
  g1[7] = 0;
  v4i z4 = {0, 0, 0, 0};
#ifdef TDM_6ARG
  v8i z8 = {0, 0, 0, 0, 0, 0, 0, 0};
  __builtin_amdgcn_tensor_load_to_lds(g0, g1, z4, z4, z8, 0);
#else
  __builtin_amdgcn_tensor_load_to_lds(g0, g1, z4, z4, 0);
#endif
}
// pad every 256 dwords (interval code 7) by 16 dwords (amount code 15)
#define W1_PADBITS ((1u << 20) | (7u << 22) | (15u << 25))
#endif

// ---------------- degree / normalization ----------------
__global__ __launch_bounds__(256) void k_init_deg(float* __restrict__ deg, int n) {
  int i = blockIdx.x * 256 + threadIdx.x;
  if (i < n) deg[i] = 1.0f;  // self-loop contribution
}

__global__ __launch_bounds__(256) void k_deg_accum(float* __restrict__ deg,
                                                   const int* __restrict__ dst, int e) {
  int i = blockIdx.x * 256 + threadIdx.x;
  if (i < e)
    __hip_atomic_fetch_add(&deg[dst[i]], 1.0f, __ATOMIC_RELAXED, __HIP_MEMORY_SCOPE_AGENT);
}

__global__ __launch_bounds__(256) void k_dinv(float* __restrict__ deg, int n) {
  int i = blockIdx.x * 256 + threadIdx.x;
  if (i < n) deg[i] = rsqrtf(deg[i]);
}

__global__ __launch_bounds__(256) void k_wn(float* __restrict__ wn,
                                            const float* __restrict__ dinv,
                                            const int* __restrict__ src,
                                            const int* __restrict__ dst, int e) {
  int i = blockIdx.x * 256 + threadIdx.x;
  if (i < e) wn[i] = dinv[src[i]] * dinv[dst[i]];
}

// ---------------- weight pre-pairing (one-time) ----------------
// W1p: [250 pair-rows][256 cols] of float2 {W1[2p][c], W1[2p+1][c]}  (contiguous, 512 f32/row)
__global__ __launch_bounds__(256) void k_pack_w1(const float* __restrict__ W1,
                                                 float* __restrict__ W1p) {
  int i = blockIdx.x * 256 + threadIdx.x;
  if (i < 250 * 256) {
    int p = i >> 8, c = i & 255;
    float2 v;
    v.x = W1[(size_t)(2 * p) * 256 + c];
    v.y = W1[(size_t)(2 * p + 1) * 256 + c];
    *(float2*)&W1p[((size_t)p * 256 + c) * 2] = v;
  }
}
// W2p: [128 pair-rows][48 col slots] float2, row stride W2ROW f32; cols >=40 zero
__global__ __launch_bounds__(256) void k_pack_w2(const float* __restrict__ W2,
                                                 float* __restrict__ W2p) {
  int i = blockIdx.x * 256 + threadIdx.x;
  if (i < 128 * 48) {
    int p = i / 48, c = i - p * 48;
    float2 v = {0.f, 0.f};
    if (c < CH) {
      v.x = W2[(size_t)(2 * p) * CH + c];
      v.y = W2[(size_t)(2 * p + 1) * CH + c];
    }
    *(float2*)&W2p[(size_t)p * W2ROW + c * 2] = v;
  }
}

// ---------------- fused MLP via f32 WMMA + TDM-staged operands ----------------
// Block: 256 threads (8 waves), 64 rows x 256 hidden cols.
// wave w: row-tile rt = w&3 (16 rows), col-half ch = w>>2 (128 cols = 8 n-tiles).
__global__ __launch_bounds__(256) void k_mlp(const float* __restrict__ x,
                                             const float* __restrict__ W1p,
                                             const float* __restrict__ b1,
                                             const float* __restrict__ W2p,
                                             const float* __restrict__ b2,
                                             float* __restrict__ h,
                                             float* __restrict__ xk) {
  __shared__ float xsl[2][ROWS * KSLAB];     // 10240 B
  __shared__ float wsl[128 * W2ROW];         // 49152 B (>= 2 * 10*W1ROW = 43520 B)
  __shared__ float h1[ROWS * H1S];           // 66560 B
  const int tid   = threadIdx.x;
  const int wave  = tid >> 5;
  const int lane  = tid & 31;
  const int mrow  = lane & 15;
  const int khalf = lane >> 4;
  const int rt    = wave & 3;
  const int ch    = wave >> 2;
  const long long rowbase = (long long)blockIdx.x * ROWS;
  const unsigned int rows_left = (unsigned int)(NNODES - rowbase);

  // slab s -> xsl[buf] (64x20) and wsl[buf region] (10 pair-rows, stride W1ROW, padded)
  auto load_slab = [&](int s, int buf) {
    const int k  = s * KSLAB;       // f32 K base
    const int ps = s * (KSLAB / 2); // pair-row base
#ifdef HAVE_TDM
    if (wave == 0) {
      tdm_load_2d((unsigned int)(size_t)&wsl[buf * 10 * W1ROW], W1p + (size_t)ps * 512,
                  512u, (unsigned int)(250 - ps), 512ull, 512u, 10u, W1_PADBITS);
      tdm_load_2d((unsigned int)(size_t)&xsl[buf][0], x + rowbase * 500 + k,
                  (unsigned int)(500 - k), rows_left, 500ull,
                  (unsigned int)KSLAB, (unsigned int)ROWS, 0u);
    }
#else
    for (int i = tid; i < 10 * 512; i += 256) {
      int lp = i >> 9, j = i & 511;
      wsl[buf * 10 * W1ROW + lp * W1ROW + j + (j >= 256 ? 16 : 0)] =
          W1p[(size_t)(ps + lp) * 512 + j];
    }
    for (int i = tid; i < ROWS * KSLAB; i += 256) {
      int r = i / KSLAB, c = i - r * KSLAB;
      float v = 0.f;
      if ((unsigned int)r < rows_left) v = x[(rowbase + r) * 500 + k + c];
      xsl[buf][i] = v;
    }
#endif
  };

  load_slab(0, 0);
#ifdef HAVE_TDM
  if (wave == 0) __builtin_amdgcn_s_wait_tensorcnt(0);
#endif
  __syncthreads();

  // ---- layer 1: 25 double-buffered slabs ----
  v8f acc[8];
#pragma unroll
  for (int t = 0; t < 8; t++) acc[t] = (v8f){0.f,0.f,0.f,0.f,0.f,0.f,0.f,0.f};
  const int arow = 16 * rt + mrow;

  for (int s = 0; s < NSLAB; s++) {
    const int buf = s & 1;
    if (s + 1 < NSLAB) load_slab(s + 1, buf ^ 1);
    const float* wbase = &wsl[buf * 10 * W1ROW + 272 * ch + 2 * mrow];
    for (int kl = 0; kl < KSLAB; kl += 4) {
      const int kk = kl + 2 * khalf;
      const int lp = (kl >> 1) + khalf;   // local pair-row 0..9
      v2f a = *(const v2f*)&xsl[buf][arow * KSLAB + kk];
      const float* wrow = wbase + lp * W1ROW;
#pragma unroll
      for (int t = 0; t < 8; t++) {
        v2f b = *(const v2f*)&wrow[32 * t];  // single aligned ds_load_b64
        acc[t] = __builtin_amdgcn_wmma_f32_16x16x4_f32(false, a, false, b, (short)0,
                                                       acc[t], false, false);
      }
    }
#ifdef HAVE_TDM
    if (wave == 0 && s + 1 < NSLAB) __builtin_amdgcn_s_wait_tensorcnt(0);
#endif
    __syncthreads();
  }

  // ---- stage W2p into wsl (reuse) overlapped with h1 epilogue ----
#ifdef HAVE_TDM
  if (wave == 0)
    tdm_load_2d((unsigned int)(size_t)&wsl[0], W2p,
                (unsigned int)W2ROW, 128u, (unsigned long long)W2ROW,
                (unsigned int)W2ROW, 128u, 0u);
#else
  for (int i = tid; i < 128 * W2ROW; i += 256) wsl[i] = W2p[i];
#endif

  // ---- bias + relu -> h1 (LDS) ----
#pragma unroll
  for (int t = 0; t < 8; t++) {
    const int col = 128 * ch + 16 * t + mrow;
    const float bias = b1[col];
#pragma unroll
    for (int v = 0; v < 8; v++) {
      const int m = 16 * rt + v + 8 * khalf;
      h1[m * H1S + col] = fmaxf(acc[t][v] + bias, 0.f);
    }
  }
#ifdef HAVE_TDM
  if (wave == 0) __builtin_amdgcn_s_wait_tensorcnt(0);
#endif
  __syncthreads();

  // ---- layer 2: 64 rows x 40 cols (3 n-tiles), K=256 all from LDS ----
  for (int t = ch; t < 3; t += 2) {       // ch=0 -> tiles 0,2 ; ch=1 -> tile 1
    v8f a2 = (v8f){0.f,0.f,0.f,0.f,0.f,0.f,0.f,0.f};
    const int  col   = t * 16 + mrow;
    const bool valid = (col < CH);
    const int  colc  = valid ? col : 0;   // clamped: garbage only feeds unstored columns
    for (int k2 = 0; k2 < 256; k2 += 4) {
      const int p2 = (k2 >> 1) + khalf;   // pair-row 0..127
      v2f a = *(const v2f*)&h1[arow * H1S + k2 + 2 * khalf];
      v2f b = *(const v2f*)&wsl[p2 * W2ROW + colc * 2];
      a2 = __builtin_amdgcn_wmma_f32_16x16x4_f32(false, a, false, b, (short)0, a2, false, false);
    }
    if (valid) {
      const float bias = b2[col];
#pragma unroll
      for (int v = 0; v < 8; v++) {
        const int m = 16 * rt + v + 8 * khalf;
        if ((unsigned int)m < rows_left) {
          const float val = a2[v] + bias;
          const long long o = (rowbase + m) * CH + col;
          h[o]  = val;
          xk[o] = val;  // xk starts as h
        }
      }
    }
  }
}

// ---------------- AMP iteration kernels ----------------
__global__ __launch_bounds__(256) void k_spmm_init(float* __restrict__ prop,
                                                   const float* __restrict__ xk,
                                                   const float* __restrict__ dinv, int total) {
  int i = blockIdx.x * 256 + threadIdx.x;
  if (i < total) {
    int node = i / CH;
    float d = dinv[node];
    prop[i] = d * d * xk[i];
  }
}

__global__ __launch_bounds__(256) void k_spmm_edges(float* __restrict__ prop,
                                                    const float* __restrict__ xk,
                                                    const float* __restrict__ wn,
                                                    const int* __restrict__ src,
                                                    const int* __restrict__ dst, int total) {
  int i = blockIdx.x * 256 + threadIdx.x;
  if (i >= total) return;
  int e = i / 10;
  int q = i - e * 10;
  float w = wn[e];
  int s = src[e], d = dst[e];
  const float4 v = *(const float4*)(xk + (size_t)s * CH + q * 4);
  float* p = prop + (size_t)d * CH + q * 4;
  __hip_atomic_fetch_add(p + 0, w * v.x, __ATOMIC_RELAXED, __HIP_MEMORY_SCOPE_AGENT);
  __hip_atomic_fetch_add(p + 1, w * v.y, __ATOMIC_RELAXED, __HIP_MEMORY_SCOPE_AGENT);
  __hip_atomic_fetch_add(p + 2, w * v.z, __ATOMIC_RELAXED, __HIP_MEMORY_SCOPE_AGENT);
  __hip_atomic_fetch_add(p + 3, w * v.w, __ATOMIC_RELAXED, __HIP_MEMORY_SCOPE_AGENT);
}

__global__ __launch_bounds__(256) void k_prox(const float* __restrict__ h,
                                              const float* __restrict__ prop,
                                              float* __restrict__ xk,
                                              float* __restrict__ score,
                                              int write_score, int n) {
  int i = blockIdx.x * 256 + threadIdx.x;
  if (i >= n) return;
  const float4* hp = (const float4*)(h    + (size_t)i * CH);
  const float4* pp = (const float4*)(prop + (size_t)i * CH);
  float4*       xp = (float4*)(xk + (size_t)i * CH);
  float4 hv[10], dv[10];
  float ss = 0.f;
#pragma unroll
  for (int q = 0; q < 10; q++) {
    float4 a = hp[q], p = pp[q];
    float4 d;
    d.x = p.x - a.x; d.y = p.y - a.y; d.z = p.z - a.z; d.w = p.w - a.w;
    ss += d.x * d.x + d.y * d.y + d.z * d.z + d.w * d.w;
    hv[q] = a; dv[q] = d;
  }
  float rn = sqrtf(ss);
  float s  = fmaxf(rn - LAM, 0.0f);
  s = (rn > 0.0f) ? (s / rn) : s;
#pragma unroll
  for (int q = 0; q < 10; q++) {
    float4 o;
    o.x = hv[q].x + s * dv[q].x; o.y = hv[q].y + s * dv[q].y;
    o.z = hv[q].z + s * dv[q].z; o.w = hv[q].w + s * dv[q].w;
    xp[q] = o;
  }
  if (write_score) score[i] = s;
}

__global__ __launch_bounds__(256) void k_logsoftmax(const float* __restrict__ xk,
                                                    float* __restrict__ out, int n) {
  int i = blockIdx.x * 256 + threadIdx.x;
  if (i >= n) return;
  const float* r = xk + (size_t)i * CH;
  float m = r[0];
#pragma unroll
  for (int c = 1; c < CH; c++) m = fmaxf(m, r[c]);
  float sum = 0.f;
#pragma unroll
  for (int c = 0; c < CH; c++) sum += expf(r[c] - m);
  float lse = logf(sum);
  float* o = out + (size_t)i * CH;
#pragma unroll
  for (int c = 0; c < CH; c++) o[c] = r[c] - m - lse;
}

// ---------------- launch ----------------
extern "C" void kernel_launch(void* const* d_in, const int* in_sizes, int n_in,
                              void* d_out, int out_size, void* d_ws, size_t ws_size,
                              hipStream_t stream) {
  const float* x  = (const float*)d_in[0];
  const float* W1 = (const float*)d_in[1];
  const float* b1 = (const float*)d_in[2];
  const float* W2 = (const float*)d_in[3];
  const float* b2 = (const float*)d_in[4];
  const int* esrc = (const int*)d_in[5];
  const int* edst = (const int*)d_in[6];
  float* out = (float*)d_out;

  float* ws   = (float*)d_ws;
  float* h    = ws;                                 // N*40
  float* xk   = ws + (size_t)NNODES * CH;           // N*40
  float* prop = ws + 2 * (size_t)NNODES * CH;       // N*40
  float* dinv = ws + 3 * (size_t)NNODES * CH;       // N
  float* wn   = dinv + NNODES;                      // E
  float* W1p  = wn + NEDGES;                        // 250*512 = 128000
  float* W2p  = W1p + 250 * 512;                    // 128*96  = 12288

  k_init_deg<<<(NNODES + 255) / 256, 256, 0, stream>>>(dinv, NNODES);
  k_deg_accum<<<(NEDGES + 255) / 256, 256, 0, stream>>>(dinv, edst, NEDGES);
  k_dinv<<<(NNODES + 255) / 256, 256, 0, stream>>>(dinv, NNODES);
  k_wn<<<(NEDGES + 255) / 256, 256, 0, stream>>>(wn, dinv, esrc, edst, NEDGES);

  k_pack_w1<<<(250 * 256 + 255) / 256, 256, 0, stream>>>(W1, W1p);
  k_pack_w2<<<(128 * 48 + 255) / 256, 256, 0, stream>>>(W2, W2p);

  k_mlp<<<(NNODES + ROWS - 1) / ROWS, 256, 0, stream>>>(x, W1p, b1, W2p, b2, h, xk);

  const int nodeC = NNODES * CH;        // 4,000,000
  const int edgeQ = NEDGES * (CH / 4);  // 16,000,000
  float* score = out + (size_t)NNODES * CH;
  for (int k = 0; k < KITER; k++) {
    k_spmm_init<<<(nodeC + 255) / 256, 256, 0, stream>>>(prop, xk, dinv, nodeC);
    k_spmm_edges<<<(edgeQ + 255) / 256, 256, 0, stream>>>(prop, xk, wn, esrc, edst, edgeQ);
    k_prox<<<(NNODES + 255) / 256, 256, 0, stream>>>(h, prop, xk, score,
                                                     (k == KITER - 1) ? 1 : 0, NNODES);
  }
  k_logsoftmax<<<(NNODES + 255) / 256, 256, 0, stream>>>(xk, out, NNODES);
}